// GenerateProposalsOp_90426241450481
// MI455X (gfx1250) — compile-verified
//
#include <hip/hip_runtime.h>
#include <math.h>
#include <stdint.h>
#include <stddef.h>

// ---------------- problem constants (from reference) ----------------
#define A_NUM 15
#define H_DIM 128
#define W_DIM 128
#define N_IMG 2
#define TOT (A_NUM * H_DIM * W_DIM)   // 245760 candidates per image
#define SORT_N 262144                 // next pow2 >= TOT
#define PRE_NMS 6000
#define POST_NMS 1000
#define KEEP_WORDS ((PRE_NMS + 31) / 32)   // 188 words of keep bits
#define NMS_THRESH 0.7f
#define BBOX_CLIP 4.135166556742356f       // log(1000/16)
#define FEAT_STRIDE 8.0f

#define CHUNK 2048                     // keys per LDS tile (16 KB of LDS)
#define CT 256                         // threads per sort block

// ---------------- CDNA5 feature detection ----------------
#if defined(__gfx1250__)
#define AS_GLOBAL __attribute__((address_space(1)))
#define AS_LOCAL  __attribute__((address_space(3)))
#if __has_builtin(__builtin_amdgcn_global_load_async_to_lds_b32)
#define HAVE_ASYNC_LDS 1
#endif
#endif

__device__ __forceinline__ void wait_async0() {
#if defined(__gfx1250__)
#if __has_builtin(__builtin_amdgcn_s_wait_asynccnt)
  __builtin_amdgcn_s_wait_asynccnt(0);
#else
  asm volatile("s_wait_asynccnt 0" ::: "memory");
#endif
#endif
}

// ---------------- stage 1: build sort keys ----------------
// key = (ordered_float_bits(score) << 32) | (0xFFFFFFFF - flat_index)
// -> descending 64-bit sort == descending score, ties broken by ascending index,
//    exactly matching jax.lax.top_k order. Pad slots get key 0 (< any real key).
__global__ void build_keys(const float* __restrict__ cls,
                           unsigned long long* __restrict__ keys, int n) {
  int t = blockIdx.x * blockDim.x + threadIdx.x;
  if (t >= SORT_N) return;
  if (t >= TOT) { keys[t] = 0ull; return; }

  float s;
#if defined(HAVE_ASYNC_LDS)
  // CDNA5 async global->LDS staging (per-lane gather into LDS), then read back.
  __shared__ float stage[256];
  float* gp = (float*)(cls + (size_t)n * TOT + t);
  __builtin_amdgcn_global_load_async_to_lds_b32(
      (AS_GLOBAL int*)gp, (AS_LOCAL int*)&stage[threadIdx.x], 0, 0);
  wait_async0();                 // per-wave; each lane reads only its own slot
  s = stage[threadIdx.x];
#else
  s = cls[(size_t)n * TOT + t];
#endif

  unsigned int bits = __float_as_uint(s);
  bits = (bits & 0x80000000u) ? ~bits : (bits | 0x80000000u);

  int w  = t % W_DIM;
  int hh = (t / W_DIM) % H_DIM;
  int a  = t / (W_DIM * H_DIM);
  int i  = (hh * W_DIM + w) * A_NUM + a;   // score-flat index (h*W + w)*A + a

  keys[i] = ((unsigned long long)bits << 32) |
            (unsigned long long)(0xFFFFFFFFu - (unsigned int)i);
}

// ---------------- stage 2: bitonic sort (descending), LDS-tiled ----------------
// Cross-chunk passes (j >= CHUNK) run in global memory (L2-resident, 2 MB array
// on a 192 MB L2). All passes with j < CHUNK run inside 16 KB LDS tiles.

// Helper: load a 2048-key chunk into LDS via the CDNA5 async engine.
// ISA: INST_OFFSET is added to BOTH the global and LDS addresses, so two b32
// async loads with offsets 0 and 4 move one 64-bit key without a VGPR bounce.
__device__ __forceinline__ void load_chunk_async(unsigned long long* __restrict__ keys,
                                                 unsigned long long* s, int base) {
#if defined(HAVE_ASYNC_LDS)
  for (int t = threadIdx.x; t < CHUNK; t += CT) {
    int* g = (int*)(keys + base + t);
    __builtin_amdgcn_global_load_async_to_lds_b32(
        (AS_GLOBAL int*)g, (AS_LOCAL int*)&s[t], 0, 0);
    __builtin_amdgcn_global_load_async_to_lds_b32(
        (AS_GLOBAL int*)g, (AS_LOCAL int*)&s[t], 4, 0);
  }
  wait_async0();       // per-wave completion
  __syncthreads();     // make all waves' async data visible block-wide
#else
  for (int t = threadIdx.x; t < CHUNK; t += CT) s[t] = keys[base + t];
  __syncthreads();
#endif
}

// Full bitonic network k=2..CHUNK within one chunk. Direction uses the GLOBAL
// index (base+i)&k: for k<CHUNK it reduces to i&k (base is CHUNK-aligned); for
// k==CHUNK it is uniform per chunk (alternating), exactly matching the global
// network's state after stage k=CHUNK.
__global__ void __launch_bounds__(CT)
bitonic_local_full(unsigned long long* __restrict__ keys) {
  __shared__ unsigned long long s[CHUNK];
  int base = blockIdx.x * CHUNK;
  load_chunk_async(keys, s, base);

  for (int k = 2; k <= CHUNK; k <<= 1) {
    for (int j = k >> 1; j > 0; j >>= 1) {
      for (int t = threadIdx.x; t < CHUNK / 2; t += CT) {
        int i = ((t & ~(j - 1)) << 1) | (t & (j - 1));  // bit log2(j) clear
        int ixj = i | j;
        bool descRegion = (((base + i) & k) == 0);
        unsigned long long a = s[i], b = s[ixj];
        if (descRegion ? (a < b) : (a > b)) { s[i] = b; s[ixj] = a; }
      }
      __syncthreads();
    }
  }
  for (int t = threadIdx.x; t < CHUNK; t += CT) keys[base + t] = s[t];
}

// Tail of stage k (j = CHUNK/2 .. 1), entirely within one chunk. For
// k >= 2*CHUNK the direction (base&k) is uniform across the chunk.
__global__ void __launch_bounds__(CT)
bitonic_local_merge(unsigned long long* __restrict__ keys, int k) {
  __shared__ unsigned long long s[CHUNK];
  int base = blockIdx.x * CHUNK;
  bool descRegion = ((base & k) == 0);
  load_chunk_async(keys, s, base);

  for (int j = CHUNK >> 1; j > 0; j >>= 1) {
    for (int t = threadIdx.x; t < CHUNK / 2; t += CT) {
      int i = ((t & ~(j - 1)) << 1) | (t & (j - 1));
      int ixj = i | j;
      unsigned long long a = s[i], b = s[ixj];
      if (descRegion ? (a < b) : (a > b)) { s[i] = b; s[ixj] = a; }
    }
    __syncthreads();
  }
  for (int t = threadIdx.x; t < CHUNK; t += CT) keys[base + t] = s[t];
}

// Cross-chunk compare-exchange pass (j >= CHUNK).
__global__ void bitonic_step(unsigned long long* __restrict__ keys, int j, int k) {
  int i = blockIdx.x * blockDim.x + threadIdx.x;
  int ixj = i ^ j;
  if (ixj <= i) return;
  unsigned long long a = keys[i];
  unsigned long long b = keys[ixj];
  bool descRegion = ((i & k) == 0);
  if (descRegion ? (a < b) : (a > b)) {
    keys[i] = b;
    keys[ixj] = a;
  }
}

// ---------------- stage 3: box decode / clip / validity ----------------
__global__ void transform_boxes(const unsigned long long* __restrict__ keys,
                                const float* __restrict__ bbox,     // image base
                                const float* __restrict__ anchors,  // 15x4
                                const float* __restrict__ im,       // [h,w,scale]
                                float* __restrict__ ox1, float* __restrict__ oy1,
                                float* __restrict__ ox2, float* __restrict__ oy2,
                                float* __restrict__ oarea,
                                float* __restrict__ oscore,
                                int* __restrict__ ovalid) {
  int r = blockIdx.x * blockDim.x + threadIdx.x;
  if (r >= PRE_NMS) return;

  unsigned long long key = keys[r];
  unsigned int idx = 0xFFFFFFFFu - (unsigned int)(key & 0xFFFFFFFFull);
  unsigned int sb = (unsigned int)(key >> 32);
  unsigned int obits = (sb & 0x80000000u) ? (sb ^ 0x80000000u) : ~sb;
  float score = __uint_as_float(obits);

  int a = (int)(idx % A_NUM);
  int p = (int)(idx / A_NUM);
  int h = p / W_DIM;   // reference: anchor shift row = p/H (== p/W since H==W)
  int w = p % W_DIM;

  // Reference swaps x/y in the anchor shifts (meshgrid 'ij' over (W,H));
  // with H==W that means: x-components shift by h*stride, y by w*stride.
  float a0 = anchors[a * 4 + 0] + (float)h * FEAT_STRIDE;
  float a1 = anchors[a * 4 + 1] + (float)w * FEAT_STRIDE;
  float a2 = anchors[a * 4 + 2] + (float)h * FEAT_STRIDE;
  float a3 = anchors[a * 4 + 3] + (float)w * FEAT_STRIDE;

  const size_t hw = (size_t)H_DIM * W_DIM;
  size_t dbase = (size_t)(4 * a) * hw + (size_t)h * W_DIM + w;
  float d0 = bbox[dbase + 0 * hw];
  float d1 = bbox[dbase + 1 * hw];
  float d2 = bbox[dbase + 2 * hw];
  float d3 = bbox[dbase + 3 * hw];

  float ws = a2 - a0 + 1.0f;
  float hs = a3 - a1 + 1.0f;
  float cx = a0 + 0.5f * ws;
  float cy = a1 + 0.5f * hs;
  float dw = fminf(d2, BBOX_CLIP);
  float dh = fminf(d3, BBOX_CLIP);
  float pcx = d0 * ws + cx;
  float pcy = d1 * hs + cy;
  float pw = expf(dw) * ws;
  float ph = expf(dh) * hs;

  float x1 = pcx - 0.5f * pw;
  float y1 = pcy - 0.5f * ph;
  float x2 = pcx + 0.5f * pw - 1.0f;
  float y2 = pcy + 0.5f * ph - 1.0f;

  float h_im = im[0], w_im = im[1], scale = im[2];
  x1 = fminf(fmaxf(x1, 0.0f), w_im - 1.0f);
  y1 = fminf(fmaxf(y1, 0.0f), h_im - 1.0f);
  x2 = fminf(fmaxf(x2, 0.0f), w_im - 1.0f);
  y2 = fminf(fmaxf(y2, 0.0f), h_im - 1.0f);

  float bws = x2 - x1 + 1.0f;
  float bhs = y2 - y1 + 1.0f;
  float min_sz = 0.0f * scale;  // MIN_SIZE == 0
  int valid = (bws >= min_sz) && (bhs >= min_sz) &&
              (x1 + bws * 0.5f < w_im) && (y1 + bhs * 0.5f < h_im);

  ox1[r] = x1; oy1[r] = y1; ox2[r] = x2; oy2[r] = y2;
  oarea[r] = bws * bhs;
  oscore[r] = score;
  ovalid[r] = valid;
}

// ---------------- stage 4a: parallel suppression mask ----------------
// One wave32 computes one 32-bit mask word (ballot == word, wave32-native).
__global__ void nms_mask(const float* __restrict__ x1, const float* __restrict__ y1,
                         const float* __restrict__ x2, const float* __restrict__ y2,
                         const float* __restrict__ area,
                         unsigned int* __restrict__ supmat) {
  int wb = blockIdx.x;                                // column word [0,188)
  int i  = blockIdx.y * blockDim.y + threadIdx.y;     // row
  int j  = wb * 32 + threadIdx.x;                     // column

  unsigned int bit = 0;
  if (i < PRE_NMS && j < PRE_NMS && j > i) {
    float ix1 = fmaxf(x1[i], x1[j]);
    float iy1 = fmaxf(y1[i], y1[j]);
    float ix2 = fminf(x2[i], x2[j]);
    float iy2 = fminf(y2[i], y2[j]);
    float iw = fmaxf(ix2 - ix1 + 1.0f, 0.0f);
    float ih = fmaxf(iy2 - iy1 + 1.0f, 0.0f);
    float inter = iw * ih;
    float iou = inter / (area[i] + area[j] - inter);
    bit = (iou > NMS_THRESH) ? 1u : 0u;
  }
  unsigned long long ball = __ballot((int)bit);       // wave32: low 32 bits
  if (threadIdx.x == 0 && i < PRE_NMS)
    supmat[(size_t)i * KEEP_WORDS + wb] = (unsigned int)ball;
}

// ---------------- stage 4b: sequential greedy reduce ----------------
__global__ void __launch_bounds__(256)
nms_reduce(const unsigned int* __restrict__ supmat,
           const int* __restrict__ valid,
           unsigned int* __restrict__ keep_out) {
  __shared__ unsigned int keepw[KEEP_WORDS];
  __shared__ int bkeep;
  int tid = threadIdx.x;

  for (int w = tid; w < KEEP_WORDS; w += 256) {
    unsigned int m = 0;
    for (int b = 0; b < 32; ++b) {
      int i = w * 32 + b;
      if (i < PRE_NMS && valid[i]) m |= (1u << b);
    }
    keepw[w] = m;
  }
  __syncthreads();

  for (int i = 0; i < PRE_NMS; ++i) {
    // Hide L2 latency of the next suppression row (global_prefetch_b8).
    if (i + 1 < PRE_NMS && tid < 12)
      __builtin_prefetch(supmat + (size_t)(i + 1) * KEEP_WORDS + tid * 16, 0, 0);
    if (tid == 0) bkeep = (int)((keepw[i >> 5] >> (i & 31)) & 1u);
    __syncthreads();
    if (bkeep) {
      const unsigned int* row = supmat + (size_t)i * KEEP_WORDS;
      for (int w = tid; w < KEEP_WORDS; w += 256) keepw[w] &= ~row[w];
    }
    __syncthreads();
  }

  for (int w = tid; w < KEEP_WORDS; w += 256) keep_out[w] = keepw[w];
}

// ---------------- stage 5: top-1000 compaction + output ----------------
// Kept entries are already in descending-score order (index order), matching
// top_k(masked, 1000); padding rows: rois=[n,0,0,0,0], prob=-1.0.
__global__ void __launch_bounds__(1024)
write_out(const unsigned int* __restrict__ keepw,
          const float* __restrict__ x1, const float* __restrict__ y1,
          const float* __restrict__ x2, const float* __restrict__ y2,
          const float* __restrict__ score,
          float* __restrict__ out, int n) {
  __shared__ int list[POST_NMS];
  __shared__ int cnt;
  if (threadIdx.x == 0) {
    int c = 0;
    for (int i = 0; i < PRE_NMS && c < POST_NMS; ++i)
      if ((keepw[i >> 5] >> (i & 31)) & 1u) list[c++] = i;
    cnt = c;
  }
  __syncthreads();

  float* rois  = out;                                  // N*1000*5
  float* probs = out + (size_t)N_IMG * POST_NMS * 5;   // N*1000
  for (int k = threadIdx.x; k < POST_NMS; k += 1024) {
    float bx1 = 0.0f, by1 = 0.0f, bx2 = 0.0f, by2 = 0.0f, sc = -1.0f;
    if (k < cnt) {
      int i = list[k];
      bx1 = x1[i]; by1 = y1[i]; bx2 = x2[i]; by2 = y2[i]; sc = score[i];
    }
    float* r = rois + ((size_t)n * POST_NMS + k) * 5;
    r[0] = (float)n; r[1] = bx1; r[2] = by1; r[3] = bx2; r[4] = by2;
    probs[(size_t)n * POST_NMS + k] = sc;
  }
}

// ---------------- host launcher ----------------
extern "C" void kernel_launch(void* const* d_in, const int* in_sizes, int n_in,
                              void* d_out, int out_size, void* d_ws, size_t ws_size,
                              hipStream_t stream) {
  (void)in_sizes; (void)n_in; (void)out_size; (void)ws_size;
  const float* cls     = (const float*)d_in[0];  // (N, A, H, W)
  const float* bbox    = (const float*)d_in[1];  // (N, 4A, H, W)
  const float* im_info = (const float*)d_in[2];  // (N, 3)
  const float* anchors = (const float*)d_in[3];  // (15, 4)
  float* out = (float*)d_out;

  // Workspace layout (reused for each image; stream serializes):
  char* ws = (char*)d_ws;
  unsigned long long* keys = (unsigned long long*)ws;           // 2 MB
  size_t off = (size_t)SORT_N * sizeof(unsigned long long);
  float* bx1   = (float*)(ws + off); off += PRE_NMS * sizeof(float);
  float* by1   = (float*)(ws + off); off += PRE_NMS * sizeof(float);
  float* bx2   = (float*)(ws + off); off += PRE_NMS * sizeof(float);
  float* by2   = (float*)(ws + off); off += PRE_NMS * sizeof(float);
  float* barea = (float*)(ws + off); off += PRE_NMS * sizeof(float);
  float* bscr  = (float*)(ws + off); off += PRE_NMS * sizeof(float);
  int*   bval  = (int*)  (ws + off); off += PRE_NMS * sizeof(int);
  unsigned int* keepw = (unsigned int*)(ws + off); off += 256 * sizeof(unsigned int);
  unsigned int* supmat = (unsigned int*)(ws + off);             // 6000*188*4 = 4.5 MB

  for (int n = 0; n < N_IMG; ++n) {
    build_keys<<<SORT_N / 256, 256, 0, stream>>>(cls, keys, n);

    // LDS-tiled bitonic sort: 1 full-local kernel, then per stage k the
    // cross-chunk global passes (j >= CHUNK) + 1 local tail kernel.
    bitonic_local_full<<<SORT_N / CHUNK, CT, 0, stream>>>(keys);
    for (int k = CHUNK * 2; k <= SORT_N; k <<= 1) {
      for (int j = k >> 1; j >= CHUNK; j >>= 1)
        bitonic_step<<<SORT_N / 256, 256, 0, stream>>>(keys, j, k);
      bitonic_local_merge<<<SORT_N / CHUNK, CT, 0, stream>>>(keys, k);
    }

    transform_boxes<<<(PRE_NMS + 255) / 256, 256, 0, stream>>>(
        keys, bbox + (size_t)n * 4 * A_NUM * H_DIM * W_DIM, anchors,
        im_info + 3 * n, bx1, by1, bx2, by2, barea, bscr, bval);

    dim3 mb(32, 8);
    dim3 mg(KEEP_WORDS, (PRE_NMS + 7) / 8);
    nms_mask<<<mg, mb, 0, stream>>>(bx1, by1, bx2, by2, barea, supmat);

    nms_reduce<<<1, 256, 0, stream>>>(supmat, bval, keepw);

    write_out<<<1, 1024, 0, stream>>>(keepw, bx1, by1, bx2, by2, bscr, out, n);
  }
}